// TCPNet_46402826666302
// MI455X (gfx1250) — compile-verified
//
#include <hip/hip_runtime.h>
#include <hip/hip_bf16.h>

#define HP 112          // hidden padded to 7 * 16
#define HID 100
#define NG 128
#define NC 8

typedef float v2f __attribute__((ext_vector_type(2)));
typedef float v8f __attribute__((ext_vector_type(8)));

// ---------------- degree / norm ----------------
__global__ void fill_kernel(float* __restrict__ p, float v, int n) {
    int i = blockIdx.x * blockDim.x + threadIdx.x;
    if (i < n) p[i] = v;
}

__global__ void deg_edge_kernel(float* __restrict__ deg, const int* __restrict__ ei, int nE) {
    int e = blockIdx.x * blockDim.x + threadIdx.x;
    if (e < nE) atomicAdd(&deg[ei[nE + e]], 1.0f);
}

__global__ void rsqrt_kernel(float* __restrict__ d, int n) {
    int i = blockIdx.x * blockDim.x + threadIdx.x;
    if (i < n) d[i] = rsqrtf(d[i]);
}

// ---------------- weight padding [100,100] -> [112,112] ----------------
__global__ void pad_w_kernel(const float* __restrict__ Wsrc, float* __restrict__ Wdst) {
    int idx = blockIdx.x * blockDim.x + threadIdx.x;
    if (idx >= HP * HP) return;
    int k = idx / HP, j = idx % HP;
    Wdst[idx] = (k < HID && j < HID) ? Wsrc[k * HID + j] : 0.0f;
}

// ---------------- layer 1: outer product x[n] * W1[j] ----------------
__global__ void l1_kernel(const float* __restrict__ x, const float* __restrict__ W1,
                          float* __restrict__ HW, int nReal, int nPadTot) {
    int idx = blockIdx.x * blockDim.x + threadIdx.x;
    if (idx >= nPadTot) return;
    int node = idx / HP, j = idx % HP;
    HW[idx] = (node < nReal && j < HID) ? x[node] * W1[j] : 0.0f;
}

// ---------------- self-loop init of aggregation buffer ----------------
__global__ void selfloop_kernel(const float* __restrict__ HW, const float* __restrict__ dinv,
                                float* __restrict__ agg, int nReal, int nPadTot) {
    int idx = blockIdx.x * blockDim.x + threadIdx.x;
    if (idx >= nPadTot) return;
    int node = idx / HP;
    float di = (node < nReal) ? dinv[node] : 0.0f;
    agg[idx] = HW[idx] * di * di;
}

// ---------------- edge scatter: wave per edge ----------------
__global__ void edge_agg_kernel(const float* __restrict__ HW, float* __restrict__ agg,
                                const int* __restrict__ ei, const float* __restrict__ dinv,
                                int nE) {
    int t = blockIdx.x * blockDim.x + threadIdx.x;
    int e = t >> 5;
    int lane = t & 31;
    if (e >= nE) return;
    int s = ei[e];
    int d = ei[nE + e];
    float w = dinv[s] * dinv[d];
    const float* hs = HW + (size_t)s * HP;
    float* ad = agg + (size_t)d * HP;
    for (int c = lane; c < HID; c += 32)
        atomicAdd(&ad[c], hs[c] * w);
}

// ---------------- bias + optional relu ----------------
__global__ void bias_act_kernel(float* __restrict__ H, const float* __restrict__ b,
                                int nReal, int nPadTot, int doRelu) {
    int idx = blockIdx.x * blockDim.x + threadIdx.x;
    if (idx >= nPadTot) return;
    int node = idx / HP, j = idx % HP;
    float v = H[idx];
    if (node < nReal && j < HID) v += b[j];
    if (doRelu) v = fmaxf(v, 0.0f);
    H[idx] = v;
}

// ---------------- WMMA GEMM: D[NpxHP] = A[NpxHP] @ W[HPxHP] ----------------
// one wave computes one 16x16 tile using V_WMMA_F32_16X16X4_F32 (28 k-steps).
// A(16x4) layout: elem (m,k) -> lane = m + 16*(k>>1), vgpr = k&1
// B(4x16) layout: elem (k,n) -> lane = n + 16*(k>>1), vgpr = k&1
// D(16x16) layout: elem (m,n) -> lane = n + 16*(m>>3), vgpr = m&7
__global__ __launch_bounds__(32)
void gemm_wmma_kernel(const float* __restrict__ A, const float* __restrict__ W,
                      float* __restrict__ D) {
    const int tileM = blockIdx.x;
    const int tileN = blockIdx.y;
    const int lane  = threadIdx.x;
    const int lo16  = lane & 15;
    const int kp    = lane >> 4;           // K-pair select (0 or 1)
    const size_t rowA = (size_t)(tileM * 16 + lo16) * HP;
    const int colB  = tileN * 16 + lo16;

    v8f c = {0.f, 0.f, 0.f, 0.f, 0.f, 0.f, 0.f, 0.f};
    #pragma unroll 4
    for (int k0 = 0; k0 < HP; k0 += 4) {
        int ka = k0 + 2 * kp;
        v2f a, b;
        a.x = A[rowA + ka];
        a.y = A[rowA + ka + 1];
        b.x = W[(size_t)ka * HP + colB];
        b.y = W[(size_t)(ka + 1) * HP + colB];
        c = __builtin_amdgcn_wmma_f32_16x16x4_f32(
                /*neg_a=*/false, a, /*neg_b=*/false, b,
                /*c_mod=*/(short)0, c, /*reuse_a=*/false, /*reuse_b=*/false);
    }

    const int mHi = (lane >> 4) * 8;
    #pragma unroll
    for (int v = 0; v < 8; ++v)
        D[(size_t)(tileM * 16 + mHi + v) * HP + tileN * 16 + lo16] = c[v];
}

// ---------------- mean pool accumulate: wave per node ----------------
__global__ void pool_kernel(const float* __restrict__ H, const int* __restrict__ batch,
                            float* __restrict__ pooled, float* __restrict__ cnt, int n) {
    int t = blockIdx.x * blockDim.x + threadIdx.x;
    int node = t >> 5;
    int lane = t & 31;
    if (node >= n) return;
    int g = batch[node];
    const float* h = H + (size_t)node * HP;
    float* p = pooled + (size_t)g * HP;
    for (int c = lane; c < HID; c += 32)
        atomicAdd(&p[c], h[c]);
    if (lane == 0) atomicAdd(&cnt[g], 1.0f);
}

// ---------------- head: out[g,c] = (pooled[g]/cnt[g]) @ Wl + bl ----------------
__global__ void head_kernel(const float* __restrict__ pooled, const float* __restrict__ cnt,
                            const float* __restrict__ Wl, const float* __restrict__ bl,
                            float* __restrict__ out) {
    int idx = blockIdx.x * blockDim.x + threadIdx.x;
    if (idx >= NG * NC) return;
    int g = idx >> 3, c = idx & 7;
    float cn = fmaxf(cnt[g], 1.0f);
    float acc = bl[c];
    for (int j = 0; j < HID; ++j)
        acc += (pooled[(size_t)g * HP + j] / cn) * Wl[j * NC + c];
    out[idx] = acc;
}

extern "C" void kernel_launch(void* const* d_in, const int* in_sizes, int n_in,
                              void* d_out, int out_size, void* d_ws, size_t ws_size,
                              hipStream_t stream) {
    const float* x     = (const float*)d_in[0];
    const int*   ei    = (const int*)d_in[1];
    const int*   batch = (const int*)d_in[2];
    const float* W1    = (const float*)d_in[3];
    const float* b1    = (const float*)d_in[4];
    const float* W2    = (const float*)d_in[5];
    const float* b2    = (const float*)d_in[6];
    const float* W3    = (const float*)d_in[7];
    const float* b3    = (const float*)d_in[8];
    const float* Wl    = (const float*)d_in[9];
    const float* bl    = (const float*)d_in[10];
    float* out = (float*)d_out;

    const int N = in_sizes[0];          // 100000
    const int E = in_sizes[1] / 2;      // 1600000
    const int Np = ((N + 15) / 16) * 16;
    const int NpTot = Np * HP;

    // carve workspace (256B aligned)
    char* ws = (char*)d_ws;
    size_t off = 0;
    auto carve = [&](size_t bytes) -> float* {
        off = (off + 255) & ~(size_t)255;
        float* p = (float*)(ws + off);
        off += bytes;
        return p;
    };
    float* dinv   = carve((size_t)N * 4);
    float* B0     = carve((size_t)NpTot * 4);
    float* B1     = carve((size_t)NpTot * 4);
    float* W2p    = carve((size_t)HP * HP * 4);
    float* W3p    = carve((size_t)HP * HP * 4);
    float* pooled = carve((size_t)NG * HP * 4);
    float* cnt    = carve((size_t)NG * 4);
    (void)ws_size;

    const int TB = 256;
    dim3 blk(TB);
    int gN     = (N + TB - 1) / TB;
    int gE     = (E + TB - 1) / TB;
    int gNpTot = (NpTot + TB - 1) / TB;
    int gEdgeW = ((E * 32) + TB - 1) / TB;   // wave per edge
    int gNodeW = ((N * 32) + TB - 1) / TB;   // wave per node
    int gWpad  = (HP * HP + TB - 1) / TB;
    dim3 gemmGrid(Np / 16, HP / 16);

    // degrees -> dinv
    fill_kernel<<<gN, blk, 0, stream>>>(dinv, 1.0f, N);            // self-loop
    deg_edge_kernel<<<gE, blk, 0, stream>>>(dinv, ei, E);
    rsqrt_kernel<<<gN, blk, 0, stream>>>(dinv, N);

    // pad weights
    pad_w_kernel<<<gWpad, blk, 0, stream>>>(W2, W2p);
    pad_w_kernel<<<gWpad, blk, 0, stream>>>(W3, W3p);

    // ---- layer 1 ----
    l1_kernel<<<gNpTot, blk, 0, stream>>>(x, W1, B0, N, NpTot);        // HW1 -> B0
    selfloop_kernel<<<gNpTot, blk, 0, stream>>>(B0, dinv, B1, N, NpTot);
    edge_agg_kernel<<<gEdgeW, blk, 0, stream>>>(B0, B1, ei, dinv, E);
    bias_act_kernel<<<gNpTot, blk, 0, stream>>>(B1, b1, N, NpTot, 1);  // H1 -> B1

    // ---- layer 2 ----
    gemm_wmma_kernel<<<gemmGrid, dim3(32), 0, stream>>>(B1, W2p, B0);  // HW2 -> B0
    selfloop_kernel<<<gNpTot, blk, 0, stream>>>(B0, dinv, B1, N, NpTot);
    edge_agg_kernel<<<gEdgeW, blk, 0, stream>>>(B0, B1, ei, dinv, E);
    bias_act_kernel<<<gNpTot, blk, 0, stream>>>(B1, b2, N, NpTot, 1);  // H2 -> B1

    // ---- layer 3 ----
    gemm_wmma_kernel<<<gemmGrid, dim3(32), 0, stream>>>(B1, W3p, B0);  // HW3 -> B0
    selfloop_kernel<<<gNpTot, blk, 0, stream>>>(B0, dinv, B1, N, NpTot);
    edge_agg_kernel<<<gEdgeW, blk, 0, stream>>>(B0, B1, ei, dinv, E);
    bias_act_kernel<<<gNpTot, blk, 0, stream>>>(B1, b3, N, NpTot, 0);  // H3 -> B1 (no relu)

    // ---- pool + head ----
    fill_kernel<<<(NG * HP + TB - 1) / TB, blk, 0, stream>>>(pooled, 0.0f, NG * HP);
    fill_kernel<<<1, blk, 0, stream>>>(cnt, 0.0f, NG);
    pool_kernel<<<gNodeW, blk, 0, stream>>>(B1, batch, pooled, cnt, N);
    head_kernel<<<(NG * NC + TB - 1) / TB, blk, 0, stream>>>(pooled, cnt, Wl, bl, out);
}